// CrfLayer_72060961292290
// MI455X (gfx1250) — compile-verified
//
#include <hip/hip_runtime.h>

// ---------------- Problem constants ----------------
#define SEQ   524288
#define T     64
#define L     512                 // steps per chunk
#define NC    (SEQ / L)           // 1024 chunks
#define FT    64                  // feat rows per TDM tile
#define START 62
#define STOP  63
#define NEGF  (-10000.0f)
#define NINF  (-3.0e38f)

// ---------------- TDM (Tensor Data Mover) helpers ----------------
typedef __attribute__((ext_vector_type(4))) unsigned int u32x4;
typedef __attribute__((ext_vector_type(4))) int          i32x4;
typedef __attribute__((ext_vector_type(8))) int          i32x8;

__device__ __forceinline__ unsigned lds_off(const void* p) {
  // generic pointers to LDS carry the workgroup-relative offset in the low 32 bits
  return (unsigned)(unsigned long long)p;
}

// Build a 2D Tensor DMA Descriptor (D#): `rows` lines of `d0` 4-byte elements,
// global line stride `stride` elems, tensor bounds d0 x trows (OOB-safe).
__device__ __forceinline__ void tdm_build(u32x4* g0, i32x8* g1, const void* lds_ptr,
                                          const void* gaddr, unsigned d0, unsigned rows,
                                          unsigned stride, unsigned trows) {
  unsigned long long ga = (unsigned long long)gaddr;
  (*g0)[0] = 1u;                                                // count=1 (valid), user mode
  (*g0)[1] = lds_off(lds_ptr);                                  // lds_addr
  (*g0)[2] = (unsigned)(ga & 0xffffffffu);                      // global_addr[31:0]
  (*g0)[3] = (unsigned)((ga >> 32) & 0x01ffffffu) | 0x80000000u; // addr[56:32] | type=2
  (*g1)[0] = (int)(2u << 16);                                   // mask=0, data_size=2 (4B)
  (*g1)[1] = (int)((d0 & 0xffffu) << 16);                       // tensor_dim0 lo16
  (*g1)[2] = (int)(((d0 >> 16) & 0xffffu) | ((trows & 0xffffu) << 16)); // dim0 hi | dim1 lo
  (*g1)[3] = (int)(((trows >> 16) & 0xffffu) | ((d0 & 0xffffu) << 16)); // dim1 hi | tile_dim0
  (*g1)[4] = (int)(rows & 0xffffu);                             // tile_dim1=rows, tile_dim2=0
  (*g1)[5] = (int)stride;                                       // tensor_dim0_stride lo32
  (*g1)[6] = 0;
  (*g1)[7] = 0;
}

__device__ __forceinline__ void tdm_load(void* lds_dst, const void* gsrc, unsigned d0,
                                         unsigned rows, unsigned stride, unsigned trows) {
  u32x4 g0; i32x8 g1;
  tdm_build(&g0, &g1, lds_dst, gsrc, d0, rows, stride, trows);
  i32x4 z4 = (i32x4)0;
#if defined(__clang_major__) && (__clang_major__ >= 23)
  i32x8 z8 = (i32x8)0;
  __builtin_amdgcn_tensor_load_to_lds(g0, g1, z4, z4, z8, 0);
#else
  __builtin_amdgcn_tensor_load_to_lds(g0, g1, z4, z4, 0);
#endif
}

__device__ __forceinline__ void tdm_store(const void* lds_src, void* gdst, unsigned d0,
                                          unsigned rows, unsigned stride, unsigned trows) {
  u32x4 g0; i32x8 g1;
  tdm_build(&g0, &g1, lds_src, gdst, d0, rows, stride, trows);
  i32x4 z4 = (i32x4)0;
#if defined(__clang_major__) && (__clang_major__ >= 23)
  i32x8 z8 = (i32x8)0;
  __builtin_amdgcn_tensor_store_from_lds(g0, g1, z4, z4, z8, 0);
#else
  __builtin_amdgcn_tensor_store_from_lds(g0, g1, z4, z4, 0);
#endif
}

// =================================================================
// K1: per-chunk max-plus transfer matrix
//   A_0[n][p]   = trans[n][p] + feat[t0][n]
//   A_{j}[n][p] = feat[t][n] + max_k( trans[n][k] + A_{j-1}[k][p] )
// 256 threads: lane n = tid&63, 16 columns p0..p0+15 per thread.
// Result stored TRANSPOSED (P[c][p*64+n]) via one TDM store, so K2
// can DMA it straight into a conflict-free LDS layout.
// =================================================================
__global__ __launch_bounds__(256) void k1_chunk_matrix(const float* __restrict__ feats,
                                                       const float* __restrict__ trans,
                                                       float* __restrict__ P) {
  __shared__ __align__(16) float transT[T * T];     // transT[k*64+n] = trans[n][k]
  __shared__ __align__(16) float Abuf[2][T * T];    // A[k*64+p]
  __shared__ __align__(16) float ftile[2][FT * T];  // feat tiles (TDM destination)
  const int tid = threadIdx.x;
  const int c = blockIdx.x;
  const int n = tid & 63;
  const int p0 = (tid >> 6) * 16;
  const long long base = (long long)c * L * T;

  if (tid == 0) tdm_load(&ftile[0][0], feats + base, T, FT, T, SEQ);
  for (int i = tid; i < T * T; i += 256) {
    int nn = i >> 6, kk = i & 63;
    transT[kk * T + nn] = trans[i];
  }
  if (tid == 0) __builtin_amdgcn_s_wait_tensorcnt(0);
  __syncthreads();

  // init A = M_{t0}
  for (int i = tid; i < T * T; i += 256) {
    int nn = i >> 6, pp = i & 63;
    Abuf[0][i] = transT[pp * T + nn] + ftile[0][nn];
  }
  const int ntiles = L / FT;
  if (tid == 0 && ntiles > 1) tdm_load(&ftile[1][0], feats + base + FT * T, T, FT, T, SEQ);

  int cur = 0;
  for (int tile = 0; tile < ntiles; ++tile) {
    const int buf = tile & 1;
    const int r0 = (tile == 0) ? 1 : 0;
    for (int r = r0; r < FT; ++r) {
      __syncthreads();
      float acc[16];
#pragma unroll
      for (int j = 0; j < 16; ++j) acc[j] = NINF;
      const float* Ac = &Abuf[cur][0];
#pragma unroll 4
      for (int k = 0; k < T; ++k) {
        const float tk = transT[k * T + n];
        const float4* ap = (const float4*)(Ac + k * T + p0);
        float4 a;
        a = ap[0];
        acc[0] = fmaxf(acc[0], tk + a.x); acc[1] = fmaxf(acc[1], tk + a.y);
        acc[2] = fmaxf(acc[2], tk + a.z); acc[3] = fmaxf(acc[3], tk + a.w);
        a = ap[1];
        acc[4] = fmaxf(acc[4], tk + a.x); acc[5] = fmaxf(acc[5], tk + a.y);
        acc[6] = fmaxf(acc[6], tk + a.z); acc[7] = fmaxf(acc[7], tk + a.w);
        a = ap[2];
        acc[8]  = fmaxf(acc[8],  tk + a.x); acc[9]  = fmaxf(acc[9],  tk + a.y);
        acc[10] = fmaxf(acc[10], tk + a.z); acc[11] = fmaxf(acc[11], tk + a.w);
        a = ap[3];
        acc[12] = fmaxf(acc[12], tk + a.x); acc[13] = fmaxf(acc[13], tk + a.y);
        acc[14] = fmaxf(acc[14], tk + a.z); acc[15] = fmaxf(acc[15], tk + a.w);
      }
      const float fn = ftile[buf][r * T + n];
      float* An = &Abuf[cur ^ 1][n * T + p0];
#pragma unroll
      for (int j = 0; j < 16; ++j) An[j] = acc[j] + fn;
      cur ^= 1;
    }
    __syncthreads();                 // all waves done reading ftile[buf]
    if (tid == 0) {
      if (tile + 2 < ntiles) {
        tdm_load(&ftile[buf][0], feats + base + (long long)(tile + 2) * FT * T, T, FT, T, SEQ);
        __builtin_amdgcn_s_wait_tensorcnt(1);   // tile+1 complete (in-order)
      } else if (tile + 1 < ntiles) {
        __builtin_amdgcn_s_wait_tensorcnt(0);
      }
    }
  }
  __syncthreads();
  // conflict-free transpose into the free buffer, then one 16 KB DMA store
  for (int i = tid; i < T * T; i += 256) {
    int nn = i >> 6, pp = i & 63;
    Abuf[cur ^ 1][pp * T + nn] = Abuf[cur][i];
  }
  __syncthreads();
  if (tid == 0)
    tdm_store(&Abuf[cur ^ 1][0], P + (long long)c * T * T, T * T, 1, T * T, 1);
  // S_ENDPGM waits idle -> tensor store completes before kernel end
}

// =================================================================
// K2: sequential scan over chunk matrices (stored transposed) ->
//     boundary fv vectors, terminal score + best tag.
//     Double-buffered 16 KB TDM loads overlap DMA with compute.
// =================================================================
__global__ __launch_bounds__(64) void k2_boundary_scan(const float* __restrict__ trans,
                                                       const float* __restrict__ P,
                                                       float* __restrict__ bound,
                                                       float* __restrict__ out0,
                                                       int* __restrict__ bestTag) {
  __shared__ __align__(16) float PL[2][T * T];   // PL[b][p*64+n] = P_c[n][p]
  __shared__ float fvA[T], fvB[T], red[T];
  const int tid = threadIdx.x;
  fvA[tid] = (tid == START) ? 0.0f : NEGF;
  if (tid == 0) tdm_load(&PL[0][0], P, T * T, 1, T * T, 1);
  __syncthreads();
  float* fc = fvA;
  float* fx = fvB;
  for (int c = 0; c < NC; ++c) {
    if (tid == 0) {
      if (c + 1 < NC) {
        tdm_load(&PL[(c + 1) & 1][0], P + (long long)(c + 1) * T * T, T * T, 1, T * T, 1);
        __builtin_amdgcn_s_wait_tensorcnt(1);   // chunk c's matrix resident
      } else {
        __builtin_amdgcn_s_wait_tensorcnt(0);
      }
    }
    __syncthreads();
    bound[c * T + tid] = fc[tid];
    const float* Pc = &PL[c & 1][0];
    float m = NINF;
#pragma unroll 8
    for (int p = 0; p < T; ++p) m = fmaxf(m, Pc[p * T + tid] + fc[p]);
    fx[tid] = m;
    __syncthreads();                 // compute done before buffer reuse next iter
    float* tmp = fc; fc = fx; fx = tmp;
  }
  bound[NC * T + tid] = fc[tid];
  red[tid] = fc[tid] + trans[STOP * T + tid];
  __syncthreads();
  if (tid == 0) {
    float b = red[0]; int bi = 0;
    for (int i = 1; i < T; ++i) if (red[i] > b) { b = red[i]; bi = i; }  // first-max
    out0[0] = b;
    bestTag[0] = bi;
  }
}

// =================================================================
// K3: replay each chunk with its entry fv: backpointers built in LDS,
//     dumped with ONE 32 KB tensor_store_from_lds; per-chunk exit-tag
//     map via 64 parallel LDS backtracks.  64 threads (one per tag).
// =================================================================
__global__ __launch_bounds__(64) void k3_replay(const float* __restrict__ feats,
                                                const float* __restrict__ trans,
                                                const float* __restrict__ bound,
                                                unsigned char* __restrict__ bpG,
                                                unsigned char* __restrict__ Bmap) {
  __shared__ __align__(16) float transT[T * T];      // transT[p*64+n] = trans[n][p]
  __shared__ __align__(16) float ftile[2][FT * T];
  __shared__ __align__(16) unsigned char bp[L * T];  // 32 KB
  __shared__ float fvA[T], fvB[T];
  const int tid = threadIdx.x;
  const int c = blockIdx.x;
  const long long base = (long long)c * L * T;

  if (tid == 0) tdm_load(&ftile[0][0], feats + base, T, FT, T, SEQ);
  for (int i = tid; i < T * T; i += 64) {
    int nn = i >> 6, pp = i & 63;
    transT[pp * T + nn] = trans[i];
  }
  fvA[tid] = bound[c * T + tid];
  const int ntiles = L / FT;
  if (tid == 0) {
    __builtin_amdgcn_s_wait_tensorcnt(0);
    if (ntiles > 1) tdm_load(&ftile[1][0], feats + base + FT * T, T, FT, T, SEQ);
  }
  __syncthreads();

  float* fc = fvA;
  float* fx = fvB;
  for (int tile = 0; tile < ntiles; ++tile) {
    const int buf = tile & 1;
    for (int r = 0; r < FT; ++r) {
      const int t = tile * FT + r;
      float best = NINF; int bi = 0;
#pragma unroll 8
      for (int p = 0; p < T; ++p) {
        float s = fc[p] + transT[p * T + tid];
        if (s > best) { best = s; bi = p; }       // strict '>' == first-max tie-break
      }
      fx[tid] = best + ftile[buf][r * T + tid];
      bp[t * T + tid] = (unsigned char)bi;
      __syncthreads();
      float* tmp = fc; fc = fx; fx = tmp;
    }
    if (tid == 0) {
      if (tile + 2 < ntiles) {
        tdm_load(&ftile[buf][0], feats + base + (long long)(tile + 2) * FT * T, T, FT, T, SEQ);
        __builtin_amdgcn_s_wait_tensorcnt(1);
      } else if (tile + 1 < ntiles) {
        __builtin_amdgcn_s_wait_tensorcnt(0);
      }
    }
    __syncthreads();
  }
  // one DMA store of the whole backpointer chunk (LDS -> HBM)
  if (tid == 0)
    tdm_store(&bp[0], bpG + base, (L * T) / 4, 1, (L * T) / 4, 1);
  // 64 parallel backtracks through this chunk's backpointers (LDS reads only)
  int tag = tid;
  for (int t = L - 1; t >= 0; --t) tag = bp[t * T + tag];
  Bmap[c * T + tid] = (unsigned char)tag;
  // S_ENDPGM waits idle -> tensor store completes before kernel end
}

// =================================================================
// K4: 1024-step scan over chunk maps -> exit tag per chunk
//     (64 KB map DMA'd into LDS with one TDM load)
// =================================================================
__global__ __launch_bounds__(64) void k4_tag_scan(const unsigned char* __restrict__ Bmap,
                                                  const int* __restrict__ bestTag,
                                                  int* __restrict__ exitTag) {
  __shared__ __align__(16) unsigned char BL[NC * T];  // 64 KB
  const int tid = threadIdx.x;
  if (tid == 0) {
    tdm_load(&BL[0], Bmap, (NC * T) / 4, 1, (NC * T) / 4, 1);
    __builtin_amdgcn_s_wait_tensorcnt(0);
  }
  __syncthreads();
  if (tid == 0) {
    int tag = bestTag[0];
    for (int c = NC - 1; c >= 0; --c) {
      exitTag[c] = tag;
      tag = BL[c * T + tag];
    }
  }
}

// =================================================================
// K5: per-chunk path emission (TDM chunk load, LDS backtrack,
//     coalesced store)
// =================================================================
__global__ __launch_bounds__(64) void k5_emit(const unsigned char* __restrict__ bpG,
                                              const int* __restrict__ exitTag,
                                              float* __restrict__ outPath) {
  __shared__ __align__(16) unsigned char bp[L * T];   // 32 KB
  __shared__ unsigned char path[L];
  const int tid = threadIdx.x;
  const int c = blockIdx.x;
  if (tid == 0) {
    tdm_load(&bp[0], bpG + (long long)c * L * T, (L * T) / 4, 1, (L * T) / 4, 1);
    __builtin_amdgcn_s_wait_tensorcnt(0);
  }
  __syncthreads();
  if (tid == 0) {
    int tag = exitTag[c];
    for (int t = L - 1; t >= 0; --t) {
      path[t] = (unsigned char)tag;
      tag = bp[t * T + tag];
    }
  }
  __syncthreads();
  for (int t = tid; t < L; t += 64) outPath[(long long)c * L + t] = (float)path[t];
}

// =================================================================
// Host launcher
// =================================================================
extern "C" void kernel_launch(void* const* d_in, const int* in_sizes, int n_in,
                              void* d_out, int out_size, void* d_ws, size_t ws_size,
                              hipStream_t stream) {
  const float* feats = (const float*)d_in[0];   // [SEQ, 64] f32
  const float* trans = (const float*)d_in[1];   // [64, 64]  f32
  float* out = (float*)d_out;                   // out[0]=score, out[1..SEQ]=tags

  // workspace layout (bytes), all offsets 256B-aligned; ~48.4 MB total
  char* ws = (char*)d_ws;
  const size_t P_OFF     = 0;                                   // NC*64*64*4 = 16 MB (transposed)
  const size_t BOUND_OFF = P_OFF + (size_t)NC * T * T * 4;      // (NC+1)*64*4
  const size_t BP_OFF    = BOUND_OFF + (size_t)(NC + 1) * T * 4;// SEQ*64 u8 = 32 MB
  const size_t BMAP_OFF  = BP_OFF + (size_t)SEQ * T;            // NC*64 u8
  const size_t EXIT_OFF  = BMAP_OFF + (size_t)NC * T;           // NC*4
  const size_t BEST_OFF  = EXIT_OFF + (size_t)NC * 4;           // 1 int

  float*         P    = (float*)(ws + P_OFF);
  float*         bnd  = (float*)(ws + BOUND_OFF);
  unsigned char* bpG  = (unsigned char*)(ws + BP_OFF);
  unsigned char* Bmap = (unsigned char*)(ws + BMAP_OFF);
  int*           exitT= (int*)(ws + EXIT_OFF);
  int*           bestT= (int*)(ws + BEST_OFF);

  k1_chunk_matrix<<<NC, 256, 0, stream>>>(feats, trans, P);
  k2_boundary_scan<<<1, 64, 0, stream>>>(trans, P, bnd, out, bestT);
  k3_replay<<<NC, 64, 0, stream>>>(feats, trans, bnd, bpG, Bmap);
  k4_tag_scan<<<1, 64, 0, stream>>>(Bmap, bestT, exitT);
  k5_emit<<<NC, 64, 0, stream>>>(bpG, exitT, out + 1);
}